// MaskLoss_62843961475340
// MI455X (gfx1250) — compile-verified
//
#include <hip/hip_runtime.h>
#include <hip/hip_bf16.h>

#define W_DIM 1024
#define H_DIM 1024
#define NPTS 42

typedef __attribute__((ext_vector_type(2))) float v2f;
typedef __attribute__((ext_vector_type(8))) float v8f;

// One block of 256 threads (8 wave32 waves). Thread t owns batch t:
//  - 42 coordinate computations + 42 scattered gathers from distmap (independent -> high MLP)
//  - per-thread serial f32 accumulation (deterministic)
// Block reduction: 256 partials -> LDS -> wave 0 folds to 64 values -> one
// v_wmma_f32_16x16x4_f32 with B=ones (column-0 sum of D == sum of all A elements,
// layout-independent) -> lanes 0 and 16 hold column 0 -> final combine + scale.
__global__ __launch_bounds__(256) void MaskLoss_kernel(
    const float* __restrict__ predict,
    const float* __restrict__ distmap,
    float* __restrict__ out, int B) {
  __shared__ float s_part[256];
  __shared__ float s_col[2];

  const int t = threadIdx.x;
  float acc = 0.0f;

  if (t < B) {
    const float* p = predict + (size_t)t * (2 * NPTS);
    const float* dm = distmap + (size_t)t * ((size_t)W_DIM * H_DIM);
#pragma unroll
    for (int n = 0; n < NPTS; ++n) {
      // contiguous, 8B-aligned pair load
      float2 pq = *(const float2*)(p + 2 * n);
      float px = pq.x;
      float py = pq.y;
      // reference: trunc(WIDTH * (px + 1.0) * 0.5) -> int32 (values in [512,1023])
      int xi = (int)((float)W_DIM * (px + 1.0f) * 0.5f);
      int yi = (int)((float)H_DIM * (py + 1.0f) * 0.5f);
      // faithful trunc(px/px)*trunc(py/py): ==1 for nonzero px,py; NaN-propagating at 0
      float ones = truncf(px / px) * truncf(py / py);
      float v = dm[(size_t)xi * W_DIM + (size_t)yi]; // distmap[b, xi, yi]
      acc += v * ones;
    }
  }

  s_part[t] = acc;
  __syncthreads();

  if (t < 32) {
    // EXEC is all 32 lanes here for wave 0 (WMMA requirement).
    // Fold 256 partials into the 64 register-slots of a 16x4 f32 A matrix.
    float a0 = s_part[t] + s_part[t + 32] + s_part[t + 64] + s_part[t + 96];
    float a1 = s_part[t + 128] + s_part[t + 160] + s_part[t + 192] + s_part[t + 224];
    v2f A;
    A[0] = a0;
    A[1] = a1;
    v2f Bm;
    Bm[0] = 1.0f;  // all-ones B (4x16): every register slot = 1.0, so the
    Bm[1] = 1.0f;  // exact A layout is irrelevant to the column sums.
    v8f C = {};
    // 8 args: (neg_a, A, neg_b, B, c_mod, C, reuse_a, reuse_b)
    v8f D = __builtin_amdgcn_wmma_f32_16x16x4_f32(
        false, A, false, Bm, (short)0, C, false, false);
    // Column N=0 of D: lane 0 holds M=0..7 in D[0..7], lane 16 holds M=8..15.
    float cs = D[0] + D[1] + D[2] + D[3] + D[4] + D[5] + D[6] + D[7];
    if (t == 0) s_col[0] = cs;
    if (t == 16) s_col[1] = cs;
  }
  __syncthreads();

  if (t == 0) {
    float total = s_col[0] + s_col[1];
    // reference division order: sum / B / N_POINTS  (÷256 exact, ÷42 rounds once)
    out[0] = total / (float)B / (float)NPTS;
  }
}

extern "C" void kernel_launch(void* const* d_in, const int* in_sizes, int n_in,
                              void* d_out, int out_size, void* d_ws, size_t ws_size,
                              hipStream_t stream) {
  (void)n_in; (void)out_size; (void)d_ws; (void)ws_size;
  const float* predict = (const float*)d_in[0];
  const float* distmap = (const float*)d_in[1];
  float* out = (float*)d_out;
  int B = in_sizes[0] / (2 * NPTS);  // 256 for this workload (fits one block)
  MaskLoss_kernel<<<1, 256, 0, stream>>>(predict, distmap, out, B);
}